// RepLoss_74732430950764
// MI455X (gfx1250) — compile-verified
//
#include <hip/hip_runtime.h>
#include <math.h>

typedef __attribute__((ext_vector_type(2))) float v2f;
typedef __attribute__((ext_vector_type(8))) float v8f;
typedef int v4i __attribute__((ext_vector_type(4)));
typedef __attribute__((address_space(1))) v4i g_v4i;   // global-AS int4
typedef __attribute__((address_space(3))) v4i l_v4i;   // LDS-AS int4

#define B_ 16
#define N_ 4096
#define G_ 256
#define M_ 65536
#define SIGMA_ 0.9f
#define EPS_ 1e-6f
#define CH_ 512          // N-chunk staged in LDS for the WMMA kernel

// ws layout (floats): [0]=iou_sum [1]=repgt_total [2]=repgt_count [3]=comgt_total [4]=comgt_count

__global__ void zero_ws_kernel(float* ws) {
    if (threadIdx.x < 8) ws[threadIdx.x] = 0.0f;
}

// ---------------- Part 1: aligned IoU -log loss over M pairs ----------------
__global__ void iou_kernel(const float* __restrict__ p, const float* __restrict__ t,
                           float* __restrict__ ws) {
    int i = blockIdx.x * blockDim.x + threadIdx.x;
    float v = 0.0f;
    if (i < M_) {
        const float* a = p + 4 * i;
        const float* b = t + 4 * i;
        float lx = fmaxf(a[0], b[0]), ly = fmaxf(a[1], b[1]);
        float rx = fminf(a[2], b[2]), ry = fminf(a[3], b[3]);
        float iw = fmaxf(rx - lx, 0.0f), ih = fmaxf(ry - ly, 0.0f);
        float inter = iw * ih;
        float areas = (a[2] - a[0]) * (a[3] - a[1]) + (b[2] - b[0]) * (b[3] - b[1]);
        float uni = fmaxf(areas - inter, EPS_);
        float iou = fmaxf(inter / uni, EPS_);
        v = -logf(iou);
    }
    __shared__ float red[256];
    red[threadIdx.x] = v;
    __syncthreads();
    for (int s = 128; s > 0; s >>= 1) {
        if (threadIdx.x < s) red[threadIdx.x] += red[threadIdx.x + s];
        __syncthreads();
    }
    if (threadIdx.x == 0) atomicAdd(&ws[0], red[0]);
}

// ---------------- Part 2: repgt (pairwise intersection max over G) ----------------
// grid = B_ * (N_/256), block = 256.  target[b] staged in LDS (4 KB), VALU-bound scan.
__global__ void repgt_kernel(const float* __restrict__ pred, const int* __restrict__ assigned,
                             const float* __restrict__ target, float* __restrict__ ws) {
    __shared__ float tg[G_ * 4];
    int b = blockIdx.x >> 4;      // N_/256 == 16 blocks per batch
    int nblk = blockIdx.x & 15;
    int tid = threadIdx.x;
    for (int i = tid; i < G_ * 4; i += 256) tg[i] = target[b * G_ * 4 + i];
    __syncthreads();

    int n = nblk * 256 + tid;
    const float* pb = pred + (size_t)(b * N_ + n) * 4;
    float px0 = pb[0], py0 = pb[1], px1 = pb[2], py1 = pb[3];
    int ag = assigned[b * N_ + n];

    float best = 0.0f;
    int arg = 0;
    for (int g = 0; g < G_; g++) {
        float tx0 = tg[g * 4 + 0], ty0 = tg[g * 4 + 1];
        float tx1 = tg[g * 4 + 2], ty1 = tg[g * 4 + 3];
        float iw = fminf(px1, tx1) - fmaxf(px0, tx0);
        float ih = fminf(py1, ty1) - fmaxf(py0, ty0);
        float ov = fmaxf(iw, 0.0f) * fmaxf(ih, 0.0f);
        ov = (g == ag) ? 0.0f : ov;
        if (ov > best) { best = ov; arg = g; }   // strict > keeps first max (matches argmax)
    }

    float val = 0.0f, isv = 0.0f;
    if (best > 0.0f) {
        float tx0 = tg[arg * 4 + 0], ty0 = tg[arg * 4 + 1];
        float tx1 = tg[arg * 4 + 2], ty1 = tg[arg * 4 + 3];
        float iw = fmaxf(fminf(px1, tx1) - fmaxf(px0, tx0), 0.0f);
        float ih = fmaxf(fminf(py1, ty1) - fmaxf(py0, ty0), 0.0f);
        float inter = iw * ih;
        float iog = inter / ((tx1 - tx0) * (ty1 - ty0));
        float sm;
        if (iog > SIGMA_)
            sm = (iog - SIGMA_) / (1.0f - SIGMA_) - logf(1.0f - SIGMA_);
        else
            sm = -logf(fmaxf(1.0f - iog, 1e-6f));
        val = sm;
        isv = 1.0f;
    }

    __shared__ float r1[256];
    __shared__ float r2[256];
    r1[tid] = val;
    r2[tid] = isv;
    __syncthreads();
    for (int s = 128; s > 0; s >>= 1) {
        if (tid < s) { r1[tid] += r1[tid + s]; r2[tid] += r2[tid + s]; }
        __syncthreads();
    }
    if (tid == 0) { atomicAdd(&ws[1], r1[0]); atomicAdd(&ws[2], r2[0]); }
}

// ---------------- Part 3: comgt via f32 WMMA ----------------
// einsum('bng,bnd->bgd'):  onehot^T (G x N) * pred_aug (N x 16-col: xyxy, ones, 0...)
// grid = B_, block = 512 (16 waves).  Chunks of N staged in LDS via CDNA5 async
// global->LDS B128 copies (ASYNCcnt-tracked), then a branchless DS-load + v_cndmask
// inner loop feeding V_WMMA_F32_16X16X4_F32 with full EXEC (exact f32 accumulation).
__global__ void comgt_wmma_kernel(const float* __restrict__ pred, const int* __restrict__ assigned,
                                  const float* __restrict__ target, float* __restrict__ ws) {
    __shared__ float s_pred[CH_ * 4];   // 8 KB pred chunk (row major)
    __shared__ int   s_asg[CH_];        // 2 KB assignment chunk
    __shared__ float tiles[16 * 256];   // 16 KB result staging (16 waves x 16x16)

    int tid  = threadIdx.x;
    int wave = tid >> 5;
    int lane = tid & 31;
    int b = blockIdx.x;
    int g0 = wave * 16;
    int m  = lane & 15;          // A: M (gt-in-tile)  /  B: N (output column)
    int kb = (lane >> 4) * 2;    // K pair base within the 4-wide K slab
    int col = (m < 4) ? m : 0;   // clamp: columns >=4 load coord 0 and get masked by select
    float colConst = (m == 4) ? 1.0f : 0.0f;   // ones column (counts); cols 5..15 -> 0

    const int* ab = assigned + b * N_;
    const float* pb = pred + (size_t)b * N_ * 4;

    v8f acc = {0.f, 0.f, 0.f, 0.f, 0.f, 0.f, 0.f, 0.f};

    for (int base = 0; base < N_; base += CH_) {
        __syncthreads();   // previous chunk fully consumed before overwrite
#if defined(__has_builtin) && __has_builtin(__builtin_amdgcn_global_load_async_to_lds_b128)
        // CDNA5 async global->LDS copy (ASYNCcnt-tracked), one B128 per lane.
        __builtin_amdgcn_global_load_async_to_lds_b128(
            (g_v4i*)(pb + (size_t)(base + tid) * 4),
            (l_v4i*)&s_pred[tid * 4], 0, 0);
        if (tid < CH_ / 4) {
            __builtin_amdgcn_global_load_async_to_lds_b128(
                (g_v4i*)(ab + base + tid * 4),
                (l_v4i*)&s_asg[tid * 4], 0, 0);
        }
#if __has_builtin(__builtin_amdgcn_s_wait_asynccnt)
        __builtin_amdgcn_s_wait_asynccnt(0);
#else
        asm volatile("s_wait_asynccnt 0x0" ::: "memory");
#endif
#else
        // fallback: vector load/store through VGPRs
        float4 pv = ((const float4*)(pb + (size_t)base * 4))[tid];
        ((float4*)s_pred)[tid] = pv;
        s_asg[tid] = ab[base + tid];
#endif
        __syncthreads();

        for (int n0 = 0; n0 < CH_; n0 += 4) {
            int r0 = n0 + kb;
            int a0 = s_asg[r0];
            int a1 = s_asg[r0 + 1];
            float p0 = s_pred[r0 * 4 + col];         // unconditional ds_load
            float p1 = s_pred[(r0 + 1) * 4 + col];
            v2f av;
            av.x = (a0 == g0 + m) ? 1.0f : 0.0f;     // A[m][kb]   (v_cndmask)
            av.y = (a1 == g0 + m) ? 1.0f : 0.0f;     // A[m][kb+1]
            v2f bv;
            bv.x = (m < 4) ? p0 : colConst;          // branchless column select
            bv.y = (m < 4) ? p1 : colConst;
            // D = A(16x4, onehot) * B(4x16, pred_aug) + C  — exact f32 accumulation
            acc = __builtin_amdgcn_wmma_f32_16x16x4_f32(
                /*neg_a=*/false, av, /*neg_b=*/false, bv,
                /*c_mod=*/(short)0, acc, /*reuse_a=*/false, /*reuse_b=*/false);
        }
    }

    // Stage D tile to LDS per the 32-bit 16x16 C/D layout:
    // VGPR v, lanes 0-15 -> row v, lanes 16-31 -> row v+8; col = lane&15.
    float* tile = tiles + wave * 256;
    int rowbase = (lane >> 4) * 8;
#pragma unroll
    for (int v = 0; v < 8; v++) tile[(rowbase + v) * 16 + (lane & 15)] = acc[v];
    __syncthreads();

    if (lane < 16) {
        float cnt = tile[lane * 16 + 4];             // ones column = count
        const float* tg = target + (size_t)(b * G_ + g0 + lane) * 4;
        float sl1 = 0.0f;
#pragma unroll
        for (int c = 0; c < 4; c++) {
            float mean = tile[lane * 16 + c] / fmaxf(cnt, 1.0f);
            float d = tg[c] - mean;
            float ad = fabsf(d);
            sl1 += (ad < 1.0f) ? 0.5f * d * d : (ad - 0.5f);
        }
        sl1 *= 0.25f;
        if (cnt > 1.0f) {
            atomicAdd(&ws[3], sl1);
            atomicAdd(&ws[4], 1.0f);
        }
    }
}

// ---------------- finalize ----------------
__global__ void finalize_kernel(const float* __restrict__ ws, float* __restrict__ out) {
    float loss = ws[0] * (1.0f / (float)M_);                              // iou mean, weight 1.0
    float n = ws[2];
    loss += (n > 0.0f) ? 10.0f * ws[1] / fmaxf(n, 1.0f) : 0.0f;           // repgt
    float mcnt = ws[4];
    loss += (mcnt > 0.0f) ? 10.0f * ws[3] / fmaxf(mcnt, 1.0f) : 0.0f;     // comgt
    out[0] = loss;
}

extern "C" void kernel_launch(void* const* d_in, const int* in_sizes, int n_in,
                              void* d_out, int out_size, void* d_ws, size_t ws_size,
                              hipStream_t stream) {
    const float* pred     = (const float*)d_in[0];   // [B,N,4]
    const int*   assigned = (const int*)d_in[1];     // [B,N]
    const float* target   = (const float*)d_in[2];   // [B,G,4]
    const float* pred2    = (const float*)d_in[3];   // [M,4]
    const float* target2  = (const float*)d_in[4];   // [M,4]
    float* ws  = (float*)d_ws;
    float* out = (float*)d_out;

    hipLaunchKernelGGL(zero_ws_kernel, dim3(1), dim3(8), 0, stream, ws);
    hipLaunchKernelGGL(iou_kernel, dim3(M_ / 256), dim3(256), 0, stream, pred2, target2, ws);
    hipLaunchKernelGGL(repgt_kernel, dim3(B_ * (N_ / 256)), dim3(256), 0, stream,
                       pred, assigned, target, ws);
    hipLaunchKernelGGL(comgt_wmma_kernel, dim3(B_), dim3(512), 0, stream,
                       pred, assigned, target, ws);
    hipLaunchKernelGGL(finalize_kernel, dim3(1), dim3(1), 0, stream, ws, out);
}